// AdaptiveResidualCombinedEncoder_7284264534657
// MI455X (gfx1250) — compile-verified
//
#include <hip/hip_runtime.h>
#include <cstdint>

#define B_DIM 2048
#define T_DIM 256
#define NFC_C 64
#define H_DIM 512
#define OUT_STRIDE 34304   // 3*512 + 2*64*256
#define SPIKE_OFF 1536
#define A_PAD 516          // 16-row A tile padded to kill LDS bank conflicts

typedef __attribute__((ext_vector_type(2))) float v2f;
typedef __attribute__((ext_vector_type(4))) float f4;
typedef __attribute__((ext_vector_type(8))) float v8f;

__device__ __forceinline__ float sigmoidf_(float x) { return 1.f / (1.f + expf(-x)); }

// ---------------------------------------------------------------------------
// Kernel 1: spikes channel soft-shift + gain, streamed through LDS via the
// CDNA5 async-to-LDS path; writes adapted_sp (NT) + per-channel summary sums.
// ---------------------------------------------------------------------------
__global__ __launch_bounds__(256) void spikes_kernel(
    const float* __restrict__ spikes, const float* __restrict__ spec_off,
    const float* __restrict__ spec_g, float* __restrict__ out,
    float* __restrict__ summary)
{
    __shared__ __align__(16) float slab[NFC_C * T_DIM];   // 64 KB: one (b,ear) slab
    __shared__ float s_lo[NFC_C], s_fr[NFC_C], s_g[NFC_C], csum[NFC_C];

    const int slabIdx = blockIdx.x;        // b*2 + ear
    const int b   = slabIdx >> 1;
    const int ear = slabIdx & 1;
    const int tid = threadIdx.x;

    const float* __restrict__ src = spikes + (size_t)slabIdx * (NFC_C * T_DIM);
    const uint64_t saddr   = (uint64_t)(uintptr_t)src;
    const uint32_t ldsBase = (uint32_t)(uintptr_t)(&slab[0]);

    // 64 KB slab -> LDS: 4096 x b128 async transfers, 16 per thread (ASYNCcnt).
    #pragma unroll
    for (int i = 0; i < 16; ++i) {
        uint32_t byteoff = (uint32_t)(tid + i * 256) * 16u;
        uint32_t ldsaddr = ldsBase + byteoff;
        asm volatile("global_load_async_to_lds_b128 %0, %1, %2"
                     :: "v"(ldsaddr), "v"(byteoff), "s"(saddr) : "memory");
    }

    // Per-channel shift params while the DMA is in flight.
    if (tid < NFC_C) {
        float s  = fminf(fmaxf((float)tid + 1.5f * tanhf(spec_off[tid]), 0.f), 63.f);
        float lo = floorf(s);
        s_lo[tid] = lo;
        s_fr[tid] = s - lo;
        s_g[tid]  = 1.f + 0.35f * tanhf(spec_g[tid]);
        csum[tid] = 0.f;
    }
    asm volatile("s_wait_asynccnt 0" ::: "memory");
    __syncthreads();

    const int cq = tid >> 6;     // 4 channels in flight per iteration
    const int t4 = tid & 63;     // 64 threads x float4 cover T=256
    float* __restrict__ outrow =
        out + (size_t)b * OUT_STRIDE + SPIKE_OFF + (size_t)ear * (NFC_C * T_DIM);

    #pragma unroll 4
    for (int it = 0; it < 16; ++it) {
        int c  = it * 4 + cq;
        int lo = (int)s_lo[c];
        int hi = lo + 1; if (hi > 63) hi = 63;
        float fr = s_fr[c], g = s_g[c];
        const f4 vlo = *(const f4*)&slab[lo * T_DIM + t4 * 4];
        const f4 vhi = *(const f4*)&slab[hi * T_DIM + t4 * 4];
        f4 v;
        v.x = (vlo.x + fr * (vhi.x - vlo.x)) * g;
        v.y = (vlo.y + fr * (vhi.y - vlo.y)) * g;
        v.z = (vlo.z + fr * (vhi.z - vlo.z)) * g;
        v.w = (vlo.w + fr * (vhi.w - vlo.w)) * g;
        __builtin_nontemporal_store(v, (f4*)&outrow[c * T_DIM + t4 * 4]);  // 256MB stream-out, never re-read
        float ps = v.x + v.y + v.z + v.w;
        #pragma unroll
        for (int off = 16; off; off >>= 1) ps += __shfl_down(ps, off, 32); // wave32
        if ((tid & 31) == 0) atomicAdd(&csum[c], ps);                      // ds_add_f32
    }
    __syncthreads();
    if (tid < NFC_C) summary[b * 128 + ear * NFC_C + tid] = csum[tid];
}

// ---------------------------------------------------------------------------
// Kernel 2: build pre-scaled concatenated weights + fused biases in ws.
// ---------------------------------------------------------------------------
__global__ __launch_bounds__(256) void wprep_kernel(
    const float* __restrict__ Wd,  const float* __restrict__ bd,
    const float* __restrict__ Wa,  const float* __restrict__ ba,
    const float* __restrict__ We,  const float* __restrict__ be,
    const float* __restrict__ Wsp, const float* __restrict__ bsp,
    const float* __restrict__ bWd, const float* __restrict__ bbd,
    const float* __restrict__ bWa, const float* __restrict__ bba,
    const float* __restrict__ bWe, const float* __restrict__ bbe,
    const float* __restrict__ dist_gain, const float* __restrict__ az_gain,
    const float* __restrict__ el_gain,
    float* __restrict__ Wcat, float* __restrict__ bcat)
{
    int idx = blockIdx.x * 256 + threadIdx.x;   // 0 .. 3*512*512-1
    int branch = idx >> 18;
    int k = (idx >> 9) & 511;
    int n = idx & 511;
    float w, bias;
    if (branch == 0) {
        float sc = 0.35f * sigmoidf_(dist_gain[0]);
        w = (k < 256) ? bWd[k * 512 + n] : sc * Wd[(k - 256) * 512 + n];
        bias = bbd[n] + sc * bd[n];
    } else if (branch == 1) {
        float sc = 0.35f * sigmoidf_(az_gain[0]);
        w = (k < 256) ? bWa[k * 512 + n] : sc * Wa[(k - 256) * 512 + n];
        bias = bba[n] + sc * ba[n];
    } else {
        float sc = 0.35f * sigmoidf_(el_gain[0]);
        w = (k < 192) ? bWe[k * 512 + n]
          : (k < 384) ? sc * We[(k - 192) * 512 + n]
                      : 0.25f * sc * Wsp[(k - 384) * 512 + n];
        bias = bbe[n] + sc * be[n] + 0.25f * sc * bsp[n];
    }
    Wcat[idx] = w;
    if (k == 0) bcat[branch * 512 + n] = bias;
}

// ---------------------------------------------------------------------------
// Block reduction (256 threads, 8 x wave32).
// ---------------------------------------------------------------------------
__device__ __forceinline__ float blockReduceSum256(float v, float* sbuf) {
    #pragma unroll
    for (int off = 16; off; off >>= 1) v += __shfl_down(v, off, 32);
    if ((threadIdx.x & 31) == 0) sbuf[threadIdx.x >> 5] = v;
    __syncthreads();
    float r = sbuf[0] + sbuf[1] + sbuf[2] + sbuf[3] +
              sbuf[4] + sbuf[5] + sbuf[6] + sbuf[7];
    __syncthreads();
    return r;
}

// ---------------------------------------------------------------------------
// Kernel 3: soft-shift + gain + layernorm of the small feature branches;
// emits the three concatenated GEMM inputs Xd/Xa/Xe ([B,512] each).
// ---------------------------------------------------------------------------
__global__ __launch_bounds__(256) void xprep_kernel(
    const float* __restrict__ distance, const float* __restrict__ azimuth,
    const float* __restrict__ elevation,
    const float* __restrict__ d_left_off, const float* __restrict__ d_right_off,
    const float* __restrict__ d_left_g,   const float* __restrict__ d_right_g,
    const float* __restrict__ az_itd_off, const float* __restrict__ az_itd_g,
    const float* __restrict__ az_ild_g,
    const float* __restrict__ el_norm_off, const float* __restrict__ el_notch_off,
    const float* __restrict__ el_slope_off, const float* __restrict__ el_norm_g,
    const float* __restrict__ el_notch_g,  const float* __restrict__ el_slope_g,
    const float* __restrict__ summary,
    float* __restrict__ Xd, float* __restrict__ Xa, float* __restrict__ Xe)
{
    __shared__ float feat[256];
    __shared__ float red[8];
    const int b = blockIdx.x, tid = threadIdx.x;

    // ---------------- distance branch ----------------
    feat[tid] = distance[b * 256 + tid];
    __syncthreads();
    float av;
    {
        int half = tid >> 7, j = tid & 127;
        const float* off = half ? d_right_off : d_left_off;
        const float* gg  = half ? d_right_g   : d_left_g;
        float s = fminf(fmaxf((float)j + 0.75f * tanhf(off[j]), 0.f), 127.f);
        int lo = (int)s; float a = s - (float)lo; int hi = lo + 1; if (hi > 127) hi = 127;
        int base = half * 128;
        float flo = feat[base + lo], fhi = feat[base + hi];
        av = (flo + a * (fhi - flo)) * (1.f + 0.35f * tanhf(gg[j]));
    }
    float mean = blockReduceSum256(av, red) * (1.f / 256.f);
    float d0 = av - mean;
    float var = blockReduceSum256(d0 * d0, red) * (1.f / 256.f);
    Xd[b * 512 + tid]       = feat[tid];
    Xd[b * 512 + 256 + tid] = d0 * rsqrtf(var + 1e-5f);
    __syncthreads();

    // ---------------- azimuth branch ----------------
    feat[tid] = azimuth[b * 256 + tid];
    __syncthreads();
    float aav;
    if (tid < 128) {
        float s = fminf(fmaxf((float)tid + 0.75f * tanhf(az_itd_off[tid]), 0.f), 127.f);
        int lo = (int)s; float a = s - (float)lo; int hi = lo + 1; if (hi > 127) hi = 127;
        float flo = feat[lo], fhi = feat[hi];
        aav = (flo + a * (fhi - flo)) * (1.f + 0.35f * tanhf(az_itd_g[tid]));
    } else {
        aav = feat[tid] * (1.f + 0.35f * tanhf(az_ild_g[tid - 128]));
    }
    mean = blockReduceSum256(aav, red) * (1.f / 256.f);
    d0 = aav - mean;
    var = blockReduceSum256(d0 * d0, red) * (1.f / 256.f);
    Xa[b * 512 + tid]       = feat[tid];
    Xa[b * 512 + 256 + tid] = d0 * rsqrtf(var + 1e-5f);
    __syncthreads();

    // ---------------- elevation branch ----------------
    if (tid < 192) feat[tid] = elevation[b * 192 + tid];
    __syncthreads();
    float ev = 0.f;
    if (tid < 192) {
        int blk = tid >> 6, j = tid & 63;
        const float* off = (blk == 0) ? el_norm_off : (blk == 1) ? el_notch_off : el_slope_off;
        const float* gg  = (blk == 0) ? el_norm_g   : (blk == 1) ? el_notch_g   : el_slope_g;
        float s = fminf(fmaxf((float)j + 1.5f * tanhf(off[j]), 0.f), 63.f);
        int lo = (int)s; float a = s - (float)lo; int hi = lo + 1; if (hi > 63) hi = 63;
        int base = blk * 64;
        float flo = feat[base + lo], fhi = feat[base + hi];
        ev = (flo + a * (fhi - flo)) * (1.f + 0.35f * tanhf(gg[j]));
    }
    float sume = blockReduceSum256((tid < 192) ? ev : 0.f, red);
    float meane = sume * (1.f / 192.f);
    float de = (tid < 192) ? ev - meane : 0.f;
    float vare = blockReduceSum256(de * de, red) * (1.f / 192.f);
    if (tid < 192) {
        Xe[b * 512 + tid]       = feat[tid];
        Xe[b * 512 + 192 + tid] = de * rsqrtf(vare + 1e-5f);
    }
    // summary layernorm (width 128)
    float sv = (tid < 128) ? summary[b * 128 + tid] : 0.f;
    float sums = blockReduceSum256(sv, red);
    float means = sums * (1.f / 128.f);
    float ds = (tid < 128) ? sv - means : 0.f;
    float vars = blockReduceSum256(ds * ds, red) * (1.f / 128.f);
    if (tid < 128) Xe[b * 512 + 384 + tid] = ds * rsqrtf(vars + 1e-5f);
}

// ---------------------------------------------------------------------------
// Kernel 4: fused GEMM + bias + ReLU via v_wmma_f32_16x16x4_f32.
// grid = (B/16, 512/128, 3); one 16x16 output tile per wave, K=512.
// ---------------------------------------------------------------------------
__global__ __launch_bounds__(256) void gemm_wmma_kernel(
    const float* __restrict__ Xin, const float* __restrict__ Wcat,
    const float* __restrict__ bcat, float* __restrict__ out)
{
    __shared__ __align__(16) float As[16 * A_PAD];   // 16x512 A tile, padded rows
    const int mtile  = blockIdx.x;
    const int branch = blockIdx.z;
    const int tid    = threadIdx.x;
    const float* __restrict__ X =
        Xin + (size_t)branch * (B_DIM * H_DIM) + (size_t)mtile * 16 * H_DIM;
    const float* __restrict__ W    = Wcat + (size_t)branch * (H_DIM * H_DIM);
    const float* __restrict__ bias = bcat + branch * H_DIM;

    #pragma unroll
    for (int i = 0; i < 8; ++i) {               // 2048 float4s, coalesced
        int e4 = tid + i * 256;
        int row = e4 >> 7, col4 = e4 & 127;
        *(f4*)&As[row * A_PAD + col4 * 4] = *(const f4*)&X[row * H_DIM + col4 * 4];
    }
    __syncthreads();

    const int w = tid >> 5, lane = tid & 31;
    const int half = lane >> 4, l = lane & 15;
    const int n0 = (blockIdx.y * 8 + w) * 16;

    v8f acc = {0.f, 0.f, 0.f, 0.f, 0.f, 0.f, 0.f, 0.f};
    for (int k = 0; k < H_DIM; k += 4) {
        const int ka = k + 2 * half;
        // A frag: lanes 0-15 hold K=ka..ka+1 of rows M=0..15; lanes 16-31 K=ka+2..ka+3
        v2f a;  a.x  = As[l * A_PAD + ka];        a.y  = As[l * A_PAD + ka + 1];
        // B frag: mirrored K striping, rows of W striped across the 16-lane halves
        v2f bf; bf.x = W[ka * H_DIM + n0 + l];    bf.y = W[(ka + 1) * H_DIM + n0 + l];
        acc = __builtin_amdgcn_wmma_f32_16x16x4_f32(false, a, false, bf,
                                                    (short)0, acc, false, false);
    }
    const float bv  = bias[n0 + l];
    const int   col = branch * H_DIM + n0 + l;
    const int  row0 = mtile * 16 + 8 * half;
    #pragma unroll
    for (int r = 0; r < 8; ++r) {
        float v = fmaxf(acc[r] + bv, 0.f);
        out[(size_t)(row0 + r) * OUT_STRIDE + col] = v;
    }
}

// ---------------------------------------------------------------------------
extern "C" void kernel_launch(void* const* d_in, const int* in_sizes, int n_in,
                              void* d_out, int out_size, void* d_ws, size_t ws_size,
                              hipStream_t stream) {
    const float* distance     = (const float*)d_in[0];
    const float* azimuth      = (const float*)d_in[1];
    const float* elevation    = (const float*)d_in[2];
    const float* spikes       = (const float*)d_in[3];
    const float* d_left_off   = (const float*)d_in[4];
    const float* d_right_off  = (const float*)d_in[5];
    const float* d_left_g     = (const float*)d_in[6];
    const float* d_right_g    = (const float*)d_in[7];
    const float* az_itd_off   = (const float*)d_in[8];
    const float* az_itd_g     = (const float*)d_in[9];
    const float* az_ild_g     = (const float*)d_in[10];
    const float* el_norm_off  = (const float*)d_in[11];
    const float* el_notch_off = (const float*)d_in[12];
    const float* el_slope_off = (const float*)d_in[13];
    const float* el_norm_g    = (const float*)d_in[14];
    const float* el_notch_g   = (const float*)d_in[15];
    const float* el_slope_g   = (const float*)d_in[16];
    const float* spec_off     = (const float*)d_in[17];
    const float* spec_g       = (const float*)d_in[18];
    const float* Wd  = (const float*)d_in[19];
    const float* bd  = (const float*)d_in[20];
    const float* Wa  = (const float*)d_in[21];
    const float* ba  = (const float*)d_in[22];
    const float* We  = (const float*)d_in[23];
    const float* be  = (const float*)d_in[24];
    const float* Wsp = (const float*)d_in[25];
    const float* bsp = (const float*)d_in[26];
    const float* bWd = (const float*)d_in[27];
    const float* bbd = (const float*)d_in[28];
    const float* bWa = (const float*)d_in[29];
    const float* bba = (const float*)d_in[30];
    const float* bWe = (const float*)d_in[31];
    const float* bbe = (const float*)d_in[32];
    const float* dist_gain = (const float*)d_in[33];
    const float* az_gain   = (const float*)d_in[34];
    const float* el_gain   = (const float*)d_in[35];

    float* out = (float*)d_out;
    float* ws  = (float*)d_ws;
    // Workspace layout (floats): summary | Xin(3 branches) | Wcat | bcat  (~16.8 MB)
    float* summary = ws;
    float* Xin  = ws + (size_t)B_DIM * 128;
    float* Wcat = Xin + (size_t)3 * B_DIM * 512;
    float* bcat = Wcat + (size_t)3 * 512 * 512;
    float* Xd = Xin;
    float* Xa = Xin + (size_t)B_DIM * 512;
    float* Xe = Xin + (size_t)2 * B_DIM * 512;

    spikes_kernel<<<B_DIM * 2, 256, 0, stream>>>(spikes, spec_off, spec_g, out, summary);
    wprep_kernel<<<(3 * 512 * 512) / 256, 256, 0, stream>>>(
        Wd, bd, Wa, ba, We, be, Wsp, bsp, bWd, bbd, bWa, bba, bWe, bbe,
        dist_gain, az_gain, el_gain, Wcat, bcat);
    xprep_kernel<<<B_DIM, 256, 0, stream>>>(
        distance, azimuth, elevation, d_left_off, d_right_off, d_left_g, d_right_g,
        az_itd_off, az_itd_g, az_ild_g, el_norm_off, el_notch_off, el_slope_off,
        el_norm_g, el_notch_g, el_slope_g, summary, Xd, Xa, Xe);
    gemm_wmma_kernel<<<dim3(B_DIM / 16, 4, 3), 256, 0, stream>>>(Xin, Wcat, bcat, out);
}